// Q_NN_9225589751876
// MI455X (gfx1250) — compile-verified
//
#include <hip/hip_runtime.h>

// ---------------------------------------------------------------------------
// Q-network fused kernel for MI455X (gfx1250, wave32).
//
// Math:  G = relu(Sg@We1+be1)@We2+be2            [u,i,j,10]
//        A = G[..,:2]^T R  (2x4)   B = R^T G (4x10)   D = A@B  (2x10)
//        Q = relu(relu(D@Wf1+bf1)@Wf2+bf2)@Wf3+bf3   [u,i,2]
//
// Roofline: 222 MB of input traffic -> ~9.5 us @ 23.3 TB/s. The embedding MLP
// (~3.3 GFLOP fp32) would be ~3x slower than that on the VALU pipe, so the two
// embedding GEMMs run on V_WMMA_F32_16X16X4_F32 (matrix pipe), one 16-row
// j-tile per wave, K/N zero-padded so padding is numerically exact. The
// j-contraction + 20->32->32->2 MLP tail is <1 GFLOP and runs lane-parallel
// on the VALU within the same 64-thread workgroup.
//
// This revision vs round 1:
//  * branchless Sg A-operand loads (clamped row + v_cndmask zeroing) instead
//    of exec-masked load branches
//  * ReLU via fmaxf -> v_max_num_f32 (was cmp+cndmask chains)
//  * dropped a mathematically redundant column select in the hbuf store
//  * fully unrolled 30-iteration contraction loops (ds_load_2addr pairs)
// ---------------------------------------------------------------------------

typedef __attribute__((ext_vector_type(2))) float v2f;
typedef __attribute__((ext_vector_type(8))) float v8f;

#define U_DIM 4096
#define I_DIM 64
#define J_DIM 30
#define NEMB  10
#define HALFD 2

__global__ __launch_bounds__(64) void qnn_fused_kernel(
    const float* __restrict__ Sg,  const float* __restrict__ R,
    const float* __restrict__ We1, const float* __restrict__ be1,
    const float* __restrict__ We2, const float* __restrict__ be2,
    const float* __restrict__ Wf1, const float* __restrict__ bf1,
    const float* __restrict__ Wf2, const float* __restrict__ bf2,
    const float* __restrict__ Wf3, const float* __restrict__ bf3,
    float* __restrict__ out)
{
    // LDS staging. hbuf: per-wave 16x12 transpose tile (C-layout -> A-layout).
    __shared__ float hbuf[2][16 * 12];
    __shared__ float Gbuf[32][NEMB];     // G rows (j padded to 32, j>=30 zeroed)
    __shared__ float Rbuf[J_DIM][4];
    __shared__ float ABbuf[48];          // [0..39] = B (4x10), [40..47] = A (2x4)
    __shared__ float Dbuf[HALFD * NEMB];
    __shared__ float H1[32];
    __shared__ float H2[32];

    const int ui  = blockIdx.x;          // flattened (u,i), 0..262143
    const int tid = threadIdx.x;         // 0..63
    const int t   = tid >> 5;            // wave id == j-tile id (0 or 1)
    const int wl  = tid & 31;            // lane in wave
    const int hi  = (wl >> 4) & 1;       // lane half (selects K pair / row+8)
    const int n   = wl & 15;             // B/C column index for this lane
    const int m   = wl & 15;             // A row-in-tile index for this lane

    // ---- per-lane WMMA B operands built straight from the tiny weights ----
    // B layout (4x16 fp32): VGPR0 holds K = 2*hi, VGPR1 holds K = 2*hi + 1.
    v2f b1;
    {
        const int k0 = 2 * hi, k1 = 2 * hi + 1;
        b1.x = (k0 < 3 && n < NEMB) ? We1[k0 * NEMB + n] : 0.0f;
        b1.y = (k1 < 3 && n < NEMB) ? We1[k1 * NEMB + n] : 0.0f;
    }
    v2f b2[3];
    #pragma unroll
    for (int p = 0; p < 3; ++p) {
        const int ka = p * 4 + 2 * hi, kb = ka + 1;
        b2[p].x = (ka < NEMB && n < NEMB) ? We2[ka * NEMB + n] : 0.0f;
        b2[p].y = (kb < NEMB && n < NEMB) ? We2[kb * NEMB + n] : 0.0f;
    }
    const float bias1 = (n < NEMB) ? be1[n] : 0.0f;
    const float bias2 = (n < NEMB) ? be2[n] : 0.0f;

    // ---- stage R rows into LDS (float4 per j) ----
    if (tid < J_DIM) {
        const float4 rv =
            reinterpret_cast<const float4*>(R)[(size_t)ui * J_DIM + tid];
        Rbuf[tid][0] = rv.x; Rbuf[tid][1] = rv.y;
        Rbuf[tid][2] = rv.z; Rbuf[tid][3] = rv.w;
    }

    // ---- GEMM1: [16x3(pad4)] @ We1[3(pad4)x10(pad16)]  ->  h1 (C layout) ----
    // A layout (16x4 fp32): lane holds row m, VGPR0=K(2*hi), VGPR1=K(2*hi+1).
    // Branchless: clamp row so every lane loads a valid address, then zero
    // the j>=30 pad rows / K=3 pad column with selects (no exec branching).
    const int j   = t * 16 + m;
    const bool jv = (j < J_DIM);
    const int jc  = jv ? j : (J_DIM - 1);
    const float* sgrow = Sg + ((size_t)ui * J_DIM + jc) * 3;
    const float v0 = sgrow[2 * hi];   // lo lanes: Sg[.,0]  hi lanes: Sg[.,2]
    const float v1 = sgrow[1];        // lo lanes: Sg[.,1]  (hi lanes unused)
    v2f a1;
    a1.x = jv ? v0 : 0.0f;
    a1.y = (jv && !hi) ? v1 : 0.0f;   // hi lanes' K=3 column is zero pad

    v8f c1 = {0.f, 0.f, 0.f, 0.f, 0.f, 0.f, 0.f, 0.f};
    c1 = __builtin_amdgcn_wmma_f32_16x16x4_f32(
             false, a1, false, b1, (short)0, c1, false, false);

    // bias + ReLU in C layout, spill 16x12 tile to LDS for the K-transpose.
    // Columns n>=10 of c1 are exactly 0 (zero-padded B) and bias1==0 there,
    // so relu(c1+bias1) is already 0 for the pad columns: no select needed.
    #pragma unroll
    for (int r = 0; r < 8; ++r) {
        const int row = r + 8 * hi;           // row-in-tile of this C element
        const float h = fmaxf(c1[r] + bias1, 0.0f);
        if (n < 12) hbuf[t][row * 12 + n] = h;
    }

    // ---- GEMM2: [16x10(pad12)] @ We2[10(pad12)x10(pad16)], 3 K-chunks ----
    v8f c2 = {0.f, 0.f, 0.f, 0.f, 0.f, 0.f, 0.f, 0.f};
    #pragma unroll
    for (int p = 0; p < 3; ++p) {
        const int kk = p * 4 + 2 * hi;
        v2f a2;
        a2.x = hbuf[t][m * 12 + kk];
        a2.y = hbuf[t][m * 12 + kk + 1];
        c2 = __builtin_amdgcn_wmma_f32_16x16x4_f32(
                 false, a2, false, b2[p], (short)0, c2, false, false);
    }

    // ---- G -> LDS, zeroing padded rows j>=30 ----
    #pragma unroll
    for (int r = 0; r < 8; ++r) {
        const int row = r + 8 * hi;
        const int jj  = t * 16 + row;
        const float g = c2[r] + bias2;
        if (n < NEMB) Gbuf[jj][n] = (jj < J_DIM) ? g : 0.0f;
    }
    __syncthreads();

    // ---- j-contraction: B = R^T G (40 elems), A = G2^T R (8 elems) ----
    if (tid < 48) {
        float s = 0.0f;
        if (tid < 40) {
            const int l = tid / NEMB, nn = tid % NEMB;
            #pragma unroll
            for (int jj = 0; jj < J_DIM; ++jj)
                s += Rbuf[jj][l] * Gbuf[jj][nn];
        } else {
            const int idx = tid - 40;
            const int k = idx >> 2, l = idx & 3;
            #pragma unroll
            for (int jj = 0; jj < J_DIM; ++jj)
                s += Gbuf[jj][k] * Rbuf[jj][l];
        }
        ABbuf[tid] = s;
    }
    __syncthreads();

    // ---- D = A @ B  (2x10) ----
    if (tid < HALFD * NEMB) {
        const int k = tid / NEMB, nn = tid % NEMB;
        float s = 0.0f;
        #pragma unroll
        for (int l = 0; l < 4; ++l)
            s += ABbuf[40 + k * 4 + l] * ABbuf[l * NEMB + nn];
        Dbuf[tid] = s;
    }
    __syncthreads();

    // ---- fitting MLP: 20 -> 32 -> 32 -> 2, one output element per lane ----
    if (tid < 32) {
        float s = bf1[tid];
        #pragma unroll
        for (int kk = 0; kk < HALFD * NEMB; ++kk)
            s += Dbuf[kk] * Wf1[kk * 32 + tid];
        H1[tid] = fmaxf(s, 0.0f);
    }
    __syncthreads();
    if (tid < 32) {
        float s = bf2[tid];
        #pragma unroll
        for (int mm = 0; mm < 32; ++mm)
            s += H1[mm] * Wf2[mm * 32 + tid];
        H2[tid] = fmaxf(s, 0.0f);
    }
    __syncthreads();
    if (tid < 2) {
        float s = bf3[tid];
        #pragma unroll
        for (int mm = 0; mm < 32; ++mm)
            s += H2[mm] * Wf3[mm * 2 + tid];
        out[(size_t)ui * 2 + tid] = s;
    }
}

extern "C" void kernel_launch(void* const* d_in, const int* in_sizes, int n_in,
                              void* d_out, int out_size, void* d_ws, size_t ws_size,
                              hipStream_t stream) {
    (void)in_sizes; (void)n_in; (void)out_size; (void)d_ws; (void)ws_size;
    const float* Sg  = (const float*)d_in[0];
    const float* R   = (const float*)d_in[1];
    const float* We1 = (const float*)d_in[2];
    const float* be1 = (const float*)d_in[3];
    const float* We2 = (const float*)d_in[4];
    const float* be2 = (const float*)d_in[5];
    const float* Wf1 = (const float*)d_in[6];
    const float* bf1 = (const float*)d_in[7];
    const float* Wf2 = (const float*)d_in[8];
    const float* bf2 = (const float*)d_in[9];
    const float* Wf3 = (const float*)d_in[10];
    const float* bf3 = (const float*)d_in[11];
    float* out = (float*)d_out;

    const int num_ui = U_DIM * I_DIM;   // 262144 workgroups, 64 threads each
    qnn_fused_kernel<<<dim3(num_ui), dim3(64), 0, stream>>>(
        Sg, R, We1, be1, We2, be2, Wf1, bf1, Wf2, bf2, Wf3, bf3, out);
}